// WaymoPostProcessing_33225867002349
// MI455X (gfx1250) — compile-verified
//
#include <hip/hip_runtime.h>
#include <hip/hip_bf16.h>
#include <stdint.h>

// Problem constants from the reference (D=1)
#define S_ 64
#define A_ 128
#define K_ 6
#define T_ 80

#define N_VALID  (S_ * T_ * A_)            /* 655360  */
#define N_TRAJ   (S_ * T_ * A_ * K_ * 2)   /* 7864320 */
#define N_SCORE  (S_ * A_ * K_)            /* 49152   */
#define N_STAK   (S_ * T_ * A_ * K_)       /* 3932160 */

typedef __attribute__((ext_vector_type(2))) float v2f;
typedef __attribute__((ext_vector_type(8))) float v8f;
typedef __attribute__((ext_vector_type(4))) int   v4i;

#if defined(__gfx1250__)
  #if __has_builtin(__builtin_amdgcn_global_load_async_to_lds_b128) && \
      __has_builtin(__builtin_amdgcn_s_wait_asynccnt)
    #define USE_ASYNC 1
  #else
    #define USE_ASYNC 0
  #endif
  #if __has_builtin(__builtin_amdgcn_wmma_f32_16x16x4_f32)
    #define USE_WMMA 1
  #else
    #define USE_WMMA 0
  #endif
#else
  #define USE_ASYNC 0
  #define USE_WMMA 0
#endif

// hardware v_sqrt_f32 (TRANS pipe) on device; plain sqrtf for host parse pass
#if defined(__AMDGCN__) && __has_builtin(__builtin_amdgcn_sqrtf)
  #define FAST_SQRT(x) __builtin_amdgcn_sqrtf(x)
#else
  #define FAST_SQRT(x) sqrtf(x)
#endif

#if USE_ASYNC
// Builtin expects int4* in AS1 (printed as __device__) / AS3 (__shared__).
typedef __attribute__((address_space(1))) v4i* gv4_ptr;
typedef __attribute__((address_space(3))) v4i* lv4_ptr;
__device__ __forceinline__ gv4_ptr as_global_v4(const void* p) {
  return (gv4_ptr)(uintptr_t)p;
}
__device__ __forceinline__ lv4_ptr as_lds_v4(const void* p) {
  // generic LDS pointers carry the LDS byte offset in the low 32 bits
  return (lv4_ptr)(uint32_t)(uintptr_t)p;
}
#endif

// pair index m in [0,15) <-> (k1,k2), k1<k2, packed as nibbles (m=15 -> (0,0))
#define PK1_PACK 0x0433222111100000ULL
#define PK2_PACK 0x0554543543254321ULL

// ---------------------------------------------------------------------------
// Kernel A: one wave per (scene, agent). Stage trajectory to LDS (async),
// compute 15 pairwise ADE sums via WMMA time-reduction (sqrt on TRANS pipe,
// accumulate on matrix pipe), then serial NMS + temperature softmax in lane 0.
// ---------------------------------------------------------------------------
__global__ __launch_bounds__(256) void nms_scores_kernel(
    const unsigned char* __restrict__ pred_valid,  // [S,A] bool
    const float* __restrict__ pred_conf,           // [1,S,A,6]
    const float* __restrict__ pred_pos,            // [1,S,A,6,80,2]
    const float* __restrict__ ref_type,            // [S,A,3]
    float* __restrict__ out_scores)                // [S,A,6]
{
  __shared__ float lds[8 * 992];                   // 8 waves * (960 traj + 32 scratch)
  const int wid  = threadIdx.x >> 5;
  const int lane = threadIdx.x & 31;
  const int g    = blockIdx.x * 8 + wid;           // agent id in [0, S*A)

  float* trj = &lds[wid * 992];                    // [k][t][2] floats, 960
  float* ps  = trj + 960;                          // 15 pair sums

  const float* gsrc = pred_pos + (size_t)g * (K_ * T_ * 2);

  // ---- stage 3840B trajectory into LDS: 240 x 16B chunks ----
#if USE_ASYNC
  for (int i = 0; i < 8; ++i) {
    int idx = i * 32 + lane;
    if (idx < 240) {
      __builtin_amdgcn_global_load_async_to_lds_b128(
          as_global_v4(gsrc + idx * 4), as_lds_v4(trj + idx * 4), 0, 0);
    }
  }
  __builtin_amdgcn_s_wait_asynccnt(0);
  __asm__ volatile("" ::: "memory");
#else
  for (int i = 0; i < 8; ++i) {
    int idx = i * 32 + lane;
    if (idx < 240) {
      float4 v = ((const float4*)gsrc)[idx];
      ((float4*)trj)[idx] = v;
    }
  }
#endif
  __syncthreads();

  // ---- pairwise ADE sums over T via WMMA reduction ----
#if USE_WMMA
  {
    const int m  = lane & 15;
    const int k1 = (int)((PK1_PACK >> (4 * m)) & 0xFULL);
    const int k2 = (int)((PK2_PACK >> (4 * m)) & 0xFULL);
    const float* r1 = trj + k1 * (T_ * 2);
    const float* r2 = trj + k2 * (T_ * 2);
    const int thalf = (lane >> 4) * 2;             // A layout: lanes>=16 hold K=2,3
    v8f c = {0.f, 0.f, 0.f, 0.f, 0.f, 0.f, 0.f, 0.f};
    v2f bone; bone[0] = 1.f; bone[1] = 1.f;        // B = ones => D[:,n] = row sums
    for (int t0 = 0; t0 < T_; t0 += 4) {
      int t = t0 + thalf;
      float dx0 = r1[t * 2 + 0] - r2[t * 2 + 0];
      float dy0 = r1[t * 2 + 1] - r2[t * 2 + 1];
      float dx1 = r1[t * 2 + 2] - r2[t * 2 + 2];
      float dy1 = r1[t * 2 + 3] - r2[t * 2 + 3];
      v2f a;
      a[0] = FAST_SQRT(dx0 * dx0 + dy0 * dy0);
      a[1] = FAST_SQRT(dx1 * dx1 + dy1 * dy1);
      c = __builtin_amdgcn_wmma_f32_16x16x4_f32(false, a, false, bone,
                                                (short)0, c, false, false);
    }
    // D column N=0: lane 0 holds M=0..7 (vgpr0..7), lane 16 holds M=8..15
    if (lane == 0) {
      ps[0] = c[0]; ps[1] = c[1]; ps[2] = c[2]; ps[3] = c[3];
      ps[4] = c[4]; ps[5] = c[5]; ps[6] = c[6]; ps[7] = c[7];
    } else if (lane == 16) {
      ps[8]  = c[0]; ps[9]  = c[1]; ps[10] = c[2]; ps[11] = c[3];
      ps[12] = c[4]; ps[13] = c[5]; ps[14] = c[6];
    }
  }
#else
  if (lane < 15) {
    const int k1 = (int)((PK1_PACK >> (4 * lane)) & 0xFULL);
    const int k2 = (int)((PK2_PACK >> (4 * lane)) & 0xFULL);
    const float* r1 = trj + k1 * (T_ * 2);
    const float* r2 = trj + k2 * (T_ * 2);
    float acc = 0.f;
    for (int t = 0; t < T_; ++t) {
      float dx = r1[t * 2 + 0] - r2[t * 2 + 0];
      float dy = r1[t * 2 + 1] - r2[t * 2 + 1];
      acc += FAST_SQRT(dx * dx + dy * dy);
    }
    ps[lane] = acc;
  }
#endif
  __syncthreads();

  // ---- serial NMS + temperature softmax (lane 0) ----
  if (lane == 0) {
    const float* cp = pred_conf + (size_t)g * K_;
    float cf[6];
    float mx = -3.402823e38f;
    for (int i = 0; i < 6; ++i) { cf[i] = cp[i]; mx = fmaxf(mx, cf[i]); }
    float sc[6];
    float sum = 0.f;
    for (int i = 0; i < 6; ++i) { sc[i] = expf(cf[i] - mx); sum += sc[i]; }
    float inv = 1.f / sum;
    for (int i = 0; i < 6; ++i) sc[i] *= inv;      // softmax (already sums to 1)

    const float* rt = ref_type + (size_t)g * 3;
    float th = rt[0] * 2.5f + rt[1] * 1.0f + rt[2] * 1.5f;

    // within-distance bitmask rows (diag true: dist 0 < th)
    unsigned wmask[6];
    for (int i = 0; i < 6; ++i) wmask[i] = 1u << i;
    for (int i = 0; i < 6; ++i) {
      for (int j = i + 1; j < 6; ++j) {
        int m = i * (11 - i) / 2 + (j - i - 1);
        if (ps[m] * (1.0f / (float)T_) < th) {
          wmask[i] |= 1u << j;
          wmask[j] |= 1u << i;
        }
      }
    }

    // stable descending argsort of original scores
    int order[6];
    bool used[6] = {false, false, false, false, false, false};
    for (int i = 0; i < 6; ++i) {
      int bi = 0; float bv = -3.402823e38f;
      for (int j = 0; j < 6; ++j)
        if (!used[j] && sc[j] > bv) { bv = sc[j]; bi = j; }
      used[bi] = true;
      order[i] = bi;
    }

    // sequential suppression scan (matches jax.lax.scan semantics)
    float cur[6];
    for (int i = 0; i < 6; ++i) cur[i] = sc[i];
    for (int i = 0; i < 6; ++i) {
      int k = order[i];
      float ck = cur[k];
      bool supp = false;
      for (int j = 0; j < 6; ++j)
        if ((wmask[k] >> j) & 1u) supp = supp || (cur[j] > ck);
      if (supp) cur[k] = 0.001f;
    }

    if (pred_valid[g] == 0)
      for (int i = 0; i < 6; ++i) cur[i] = sc[i];

    float s2 = 0.f;
    for (int i = 0; i < 6; ++i) s2 += cur[i];
    float inv2 = 1.f / s2;
    for (int i = 0; i < 6; ++i) cur[i] *= inv2;    // renormalize

    // softmax(log(s) / 0.5)
    float lg[6];
    float mx2 = -3.402823e38f;
    for (int i = 0; i < 6; ++i) { lg[i] = 2.0f * logf(cur[i]); mx2 = fmaxf(mx2, lg[i]); }
    float s3 = 0.f;
    float ee[6];
    for (int i = 0; i < 6; ++i) { ee[i] = expf(lg[i] - mx2); s3 += ee[i]; }
    float inv3 = 1.f / s3;

    float* op = out_scores + (size_t)g * K_;
    for (int i = 0; i < 6; ++i) op[i] = ee[i] * inv3;
  }
}

// ---------------------------------------------------------------------------
// Kernel B: layout permute, write-coalesced; strided reads hit the 192MB L2.
//   trajs: [1,S,A,K,T,2] -> [S,T,A,K,2]
//   yaw:   [1,S,A,K,T,1] -> [S,T,A,K,1]
//   valid: [S,A]         -> [S,T,A] (as 0/1 float)
// ---------------------------------------------------------------------------
__global__ __launch_bounds__(256) void permute_kernel(
    const float* __restrict__ pred_pos,
    const float* __restrict__ pred_yaw,
    const unsigned char* __restrict__ pred_valid,
    float* __restrict__ out)
{
  const int j = blockIdx.x * 256 + threadIdx.x;
  if (j < N_STAK) {
    int k = j % K_;
    int a = (j / K_) % A_;
    int t = (j / (K_ * A_)) % T_;
    int s = j / (K_ * A_ * T_);
    int in_base = ((s * A_ + a) * K_ + k) * T_ + t;     // element (float2 / float) index
    float2 p = ((const float2*)pred_pos)[in_base];
    ((float2*)(out + N_VALID))[j] = p;                  // trajs
    out[N_VALID + N_TRAJ + N_SCORE + j] = pred_yaw[in_base];  // yaw_bbox
  }
  if (j < N_VALID) {
    int a = j % A_;
    int s = j / (A_ * T_);
    out[j] = pred_valid[s * A_ + a] ? 1.0f : 0.0f;      // valid
  }
}

extern "C" void kernel_launch(void* const* d_in, const int* in_sizes, int n_in,
                              void* d_out, int out_size, void* d_ws, size_t ws_size,
                              hipStream_t stream) {
  (void)in_sizes; (void)n_in; (void)out_size; (void)d_ws; (void)ws_size;
  const unsigned char* pv = (const unsigned char*)d_in[0];  // pred_valid (bool)
  const float* conf = (const float*)d_in[1];                // pred_conf
  const float* pos  = (const float*)d_in[2];                // pred_pos
  const float* yaw  = (const float*)d_in[3];                // pred_yaw_bbox
  const float* rt   = (const float*)d_in[4];                // ref_type
  float* out = (float*)d_out;

  // Kernel A: one wave per agent -> S*A/8 blocks of 256 threads (8 waves)
  nms_scores_kernel<<<(S_ * A_) / 8, 256, 0, stream>>>(
      pv, conf, pos, rt, out + N_VALID + N_TRAJ);

  // Kernel B: permute trajs/yaw/valid
  permute_kernel<<<(N_STAK + 255) / 256, 256, 0, stream>>>(pos, yaw, pv, out);
}